// RESCALScore_1872605741813
// MI455X (gfx1250) — compile-verified
//
#include <hip/hip_runtime.h>

// RESCAL edge scoring: score[e] = head_e^T * R_e * tail_e
//   node_emb: [50000, 128] f32
//   rel_emb : [32768, 128*128] f32   (2.1 GB stream -> HBM bound, ~92us floor)
//   src/dst : [32768] int32
//   out     : [32768] f32
//
// One 256-thread block (8 waves) per edge. Wave w owns rows [16w,16w+16) of
// R_e, streams them HBM->LDS with global_load_async_to_lds_b128 (CDNA5 async
// data mover, ASYNCcnt-tracked), then runs 32x V_WMMA_F32_16X16X4_F32 with a
// masked-tail B matrix: output column n accumulates matvec dims [8n, 8n+8).
// Head-weighted reduce via ds_add_f32.

#define DIM        128
#define ROWPAD     4                 // 16B pad -> conflict-free strided ds reads
#define ROWSTRIDE  (DIM + ROWPAD)    // 132 floats = 528 bytes per row

typedef float v2f __attribute__((ext_vector_type(2)));
typedef float v8f __attribute__((ext_vector_type(8)));
typedef __attribute__((address_space(3))) float lds_f;

__global__ __launch_bounds__(256)
void rescal_score_kernel(const float* __restrict__ node_emb,
                         const float* __restrict__ rel_emb,
                         const int*   __restrict__ src,
                         const int*   __restrict__ dst,
                         float*       __restrict__ out)
{
    __shared__ __align__(16) float relS[8 * 16 * ROWSTRIDE]; // 67584 B
    __shared__ __align__(16) float tailS[DIM];
    __shared__ __align__(16) float headS[DIM];
    __shared__ float accS;

    const int e    = blockIdx.x;
    const int tid  = threadIdx.x;
    const int w    = tid >> 5;    // wave id: 0..7
    const int lane = tid & 31;

    if (tid == 0) accS = 0.0f;

    // ---- stage tail (threads 0..127) and head (threads 128..255) ----
    if (tid < DIM) {
        tailS[tid] = node_emb[(long)dst[e] * DIM + tid];
    } else {
        const int i = tid - DIM;
        headS[i] = node_emb[(long)src[e] * DIM + i];
    }

    // ---- async-stage R_e rows [16w, 16w+16): HBM -> LDS, no VGPR roundtrip --
    // Per instruction: each lane moves 16B; one wave iteration = one 512B row.
    const float* relBase = rel_emb + (long)e * (DIM * DIM);
    #pragma unroll 4
    for (int it = 0; it < 16; ++it) {
        const int row = w * 16 + it;
        const float* gp = relBase + (long)row * DIM + lane * 4;
        const unsigned ldsOff =
            (unsigned)(unsigned long long)(lds_f*)&relS[row * ROWSTRIDE + lane * 4];
        asm volatile("global_load_async_to_lds_b128 %0, %1, off"
                     :: "v"(ldsOff), "v"(gp)
                     : "memory");
    }
    asm volatile("s_wait_asynccnt 0x0" ::: "memory");
    __syncthreads();

    // ---- WMMA: O(16x16) = sum_kc A_kc(16x4) x B_kc(4x16) ----
    // f32 A/B layout: lanes 0-15 hold M(resp N)=0..15, K={0,1}; lanes 16-31
    // hold K={2,3}. B_kc[k,n] = tail[4*kc+k] iff n == (kc>>1): each lane's B
    // is nonzero only for kc in {2n, 2n+1} -> 4 hoisted tail floats per lane.
    const int n     = lane & 15;
    const int khi   = lane >> 4;      // 0 or 1
    const int kpair = khi * 2;        // K offset within 4-chunk

    const float* tp = &tailS[8 * n + kpair];
    const float tA0 = tp[0];          // kc = 2n,  K = kpair
    const float tA1 = tp[1];          // kc = 2n,  K = kpair+1
    const float tB0 = tp[4];          // kc = 2n+1
    const float tB1 = tp[5];

    const float* arowBase = &relS[(w * 16 + n) * ROWSTRIDE + kpair];

    v8f acc = {0.f, 0.f, 0.f, 0.f, 0.f, 0.f, 0.f, 0.f};

    #pragma unroll
    for (int g = 0; g < 16; ++g) {    // chunk pair: kc = 2g, 2g+1
        const bool hit = (n == g);    // pure value selects -> v_cndmask, no exec games
        v2f b0, b1;
        b0.x = hit ? tA0 : 0.0f;
        b0.y = hit ? tA1 : 0.0f;
        b1.x = hit ? tB0 : 0.0f;
        b1.y = hit ? tB1 : 0.0f;

        const float* arow = arowBase + g * 8;
        v2f a0, a1;
        a0.x = arow[0];
        a0.y = arow[1];
        a1.x = arow[4];
        a1.y = arow[5];

        acc = __builtin_amdgcn_wmma_f32_16x16x4_f32(
            false, a0, false, b0, (short)0, acc, false, false);
        acc = __builtin_amdgcn_wmma_f32_16x16x4_f32(
            false, a1, false, b1, (short)0, acc, false, false);
    }

    // ---- head-weighted reduction ----
    // C/D layout: lane<16 -> M = v; lane>=16 -> M = v + 8 (within wave tile).
    float s = 0.0f;
    #pragma unroll
    for (int v = 0; v < 8; ++v) {
        s += acc[v] * headS[w * 16 + v + khi * 8];
    }
    atomicAdd(&accS, s);
    __syncthreads();

    if (tid == 0) out[e] = accS;
}

extern "C" void kernel_launch(void* const* d_in, const int* in_sizes, int n_in,
                              void* d_out, int out_size, void* d_ws, size_t ws_size,
                              hipStream_t stream) {
    const float* node_emb = (const float*)d_in[0];
    const float* rel_emb  = (const float*)d_in[1];
    const int*   src      = (const int*)d_in[2];
    const int*   dst      = (const int*)d_in[3];
    float*       out      = (float*)d_out;

    const int E = in_sizes[2];   // 32768 edges

    rescal_score_kernel<<<E, 256, 0, stream>>>(node_emb, rel_emb, src, dst, out);
}